// FlowGNN_2259152798367
// MI455X (gfx1250) — compile-verified
//
#include <hip/hip_runtime.h>
#include <hip/hip_bf16.h>

#define NN   50000   // nodes
#define NE   800000  // edges (before self loops)
#define HID  64
#define NLAY 3

typedef __attribute__((ext_vector_type(2))) float v2f;
typedef __attribute__((ext_vector_type(8))) float v8f;

// ---------------- graph normalization ----------------
__global__ void k_init_deg(float* deg) {
    int i = blockIdx.x * blockDim.x + threadIdx.x;
    if (i < NN) deg[i] = 1.0f;  // self loop
}

__global__ void k_count_deg(const int* __restrict__ dst, float* __restrict__ deg) {
    int e = blockIdx.x * blockDim.x + threadIdx.x;
    if (e < NE) atomicAdd(&deg[dst[e]], 1.0f);
}

__global__ void k_rsqrt_deg(float* deg) {
    int i = blockIdx.x * blockDim.x + threadIdx.x;
    if (i < NN) deg[i] = rsqrtf(fmaxf(deg[i], 1e-12f));
}

__global__ void k_enorm(const int* __restrict__ src, const int* __restrict__ dst,
                        const float* __restrict__ dinv, float* __restrict__ enorm) {
    int e = blockIdx.x * blockDim.x + threadIdx.x;
    if (e < NE) {
        enorm[e] = dinv[src[e]] * dinv[dst[e]];
    } else if (e < NE + NN) {
        float d = dinv[e - NE];
        enorm[e] = d * d;
    }
}

// ---------------- input projection: h = x @ W_in + b_in (K=4) ----------------
__global__ void k_input_proj(const float* __restrict__ x, const float* __restrict__ Win,
                             const float* __restrict__ bin, float* __restrict__ h) {
    int idx = blockIdx.x * blockDim.x + threadIdx.x;   // (node, c)
    if (idx >= NN * HID) return;
    int node = idx >> 6, c = idx & 63;
    const float* xr = x + node * 4;
    float acc = bin[c];
    acc += xr[0] * Win[0 * HID + c];
    acc += xr[1] * Win[1 * HID + c];
    acc += xr[2] * Win[2 * HID + c];
    acc += xr[3] * Win[3 * HID + c];
    h[idx] = acc;
}

// ---------------- 64x64 fp32 GEMM via V_WMMA_F32_16X16X4_F32 ----------------
// grid = NN/16 blocks, block = 128 threads (4 waves). Wave w computes the
// 16x16 tile (rows = node tile, cols = [16w,16w+16)). K=64 -> 16 WMMA steps.
__global__ void __launch_bounds__(128)
k_gemm64_wmma(const float* __restrict__ A, const float* __restrict__ W,
              const float* __restrict__ bias, float* __restrict__ out, int do_relu) {
    const int lane = threadIdx.x & 31;
    const int wv   = threadIdx.x >> 5;           // column tile 0..3
    const int r0   = blockIdx.x * 16;            // node row tile (NN%16==0)
    const int mrow = r0 + (lane & 15);
    const int koff = (lane < 16) ? 0 : 2;        // A/B K-half per ISA layout
    const int n    = wv * 16 + (lane & 15);

    const float* arow = A + mrow * HID + koff;

    v8f acc = {};
#pragma unroll
    for (int k = 0; k < 16; ++k) {
        const int k0 = 4 * k;
        v2f a = *(const v2f*)(arow + k0);        // 8B aligned (even float offset)
        v2f b;
        b.x = W[(k0 + koff) * HID + n];
        b.y = W[(k0 + koff + 1) * HID + n];
        acc = __builtin_amdgcn_wmma_f32_16x16x4_f32(
            /*neg_a=*/false, a, /*neg_b=*/false, b,
            /*c_mod=*/(short)0, acc, /*reuse_a=*/false, /*reuse_b=*/false);
    }

    const float bn = bias ? bias[n] : 0.0f;
    const int mhalf = (lane >= 16) ? 8 : 0;      // C/D layout: VGPR r -> M=r (+8 hi lanes)
#pragma unroll
    for (int r = 0; r < 8; ++r) {
        float v = acc[r] + bn;
        if (do_relu) v = fmaxf(v, 0.0f);
        out[(r0 + r + mhalf) * HID + n] = v;
    }
}

// ---------------- zero buffer ----------------
__global__ void k_zero(float* __restrict__ p, int n) {
    int i = blockIdx.x * blockDim.x + threadIdx.x;
    if (i < n) p[i] = 0.0f;
}

// ---------------- edge gather + normalized scatter-add ----------------
// one thread per (edge, channel); lanes cover consecutive channels of one
// src row -> coalesced 128B gather per wave; scatter via f32 atomics (L2).
__global__ void k_scatter(const float* __restrict__ hw, const int* __restrict__ src,
                          const int* __restrict__ dst, const float* __restrict__ enorm,
                          float* __restrict__ agg) {
    long long tid = (long long)blockIdx.x * blockDim.x + threadIdx.x;
    if (tid >= (long long)(NE + NN) * HID) return;
    int e = (int)(tid >> 6);
    int c = (int)(tid & 63);
    int s, d;
    if (e < NE) { s = src[e]; d = dst[e]; }
    else        { s = d = e - NE; }
    atomicAdd(&agg[(long long)d * HID + c], hw[(long long)s * HID + c] * enorm[e]);
}

// ---------------- conv bias + BatchNorm(eval) + ReLU + residual (in place) ----------------
__global__ void k_bn_relu_res(const float* __restrict__ agg, const float* __restrict__ cb,
                              const float* __restrict__ gamma, const float* __restrict__ beta,
                              const float* __restrict__ mean, const float* __restrict__ var,
                              float* __restrict__ h) {
    int idx = blockIdx.x * blockDim.x + threadIdx.x;
    if (idx >= NN * HID) return;
    int c = idx & 63;
    float v = agg[idx] + cb[c];
    v = (v - mean[c]) * rsqrtf(var[c] + 1e-5f) * gamma[c] + beta[c];
    v = fmaxf(v, 0.0f);
    h[idx] = v + h[idx];
}

// ---------------- final 64->4 projection + autoregressive residual ----------------
__global__ void k_out_proj(const float* __restrict__ t, const float* __restrict__ W2,
                           const float* __restrict__ b2, const float* __restrict__ x,
                           float* __restrict__ out) {
    int idx = blockIdx.x * blockDim.x + threadIdx.x;   // (node, oc)
    if (idx >= NN * 4) return;
    int node = idx >> 2, oc = idx & 3;
    const float* tr = t + node * HID;
    float acc = b2[oc];
#pragma unroll
    for (int j = 0; j < HID; ++j) acc += tr[j] * W2[j * 4 + oc];
    out[idx] = x[idx] + acc;
}

extern "C" void kernel_launch(void* const* d_in, const int* in_sizes, int n_in,
                              void* d_out, int out_size, void* d_ws, size_t ws_size,
                              hipStream_t stream) {
    const float* x     = (const float*)d_in[0];
    const int*   ei    = (const int*)  d_in[1];
    const float* Win   = (const float*)d_in[2];
    const float* bin   = (const float*)d_in[3];
    const float* convw = (const float*)d_in[4];
    const float* convb = (const float*)d_in[5];
    const float* gamma = (const float*)d_in[6];
    const float* beta  = (const float*)d_in[7];
    const float* mean  = (const float*)d_in[8];
    const float* var   = (const float*)d_in[9];
    const float* W1    = (const float*)d_in[10];
    const float* b1    = (const float*)d_in[11];
    const float* W2    = (const float*)d_in[12];
    const float* b2    = (const float*)d_in[13];
    float* out = (float*)d_out;

    const int* src = ei;            // edge_index[0]
    const int* dst = ei + NE;       // edge_index[1]

    // workspace layout (floats)
    float* ws    = (float*)d_ws;
    float* dinv  = ws;                         // NN
    float* enorm = dinv + NN;                  // NE+NN
    float* h     = enorm + (NE + NN);          // NN*HID
    float* hw    = h  + (size_t)NN * HID;      // NN*HID
    float* agg   = hw + (size_t)NN * HID;      // NN*HID

    const int T = 256;
    const int gN   = (NN + T - 1) / T;
    const int gE   = (NE + T - 1) / T;
    const int gEN  = (NE + NN + T - 1) / T;
    const int gNH  = (NN * HID) / T;                         // 12500 exact
    const long long scat_total = (long long)(NE + NN) * HID; // 54.4M
    const int gScat = (int)((scat_total + T - 1) / T);
    const int gGemm = NN / 16;                               // 3125 exact

    // graph normalization
    k_init_deg <<<gN, T, 0, stream>>>(dinv);
    k_count_deg<<<gE, T, 0, stream>>>(dst, dinv);
    k_rsqrt_deg<<<gN, T, 0, stream>>>(dinv);
    k_enorm    <<<gEN, T, 0, stream>>>(src, dst, dinv, enorm);

    // input projection
    k_input_proj<<<gNH, T, 0, stream>>>(x, Win, bin, h);

    // GCN layers
    for (int l = 0; l < NLAY; ++l) {
        k_gemm64_wmma<<<gGemm, 128, 0, stream>>>(h, convw + (size_t)l * HID * HID,
                                                 nullptr, hw, 0);
        k_zero   <<<gNH, T, 0, stream>>>(agg, NN * HID);
        k_scatter<<<gScat, T, 0, stream>>>(hw, src, dst, enorm, agg);
        k_bn_relu_res<<<gNH, T, 0, stream>>>(agg, convb + l * HID, gamma + l * HID,
                                             beta + l * HID, mean + l * HID,
                                             var + l * HID, h);
    }

    // output MLP: relu(h@W1+b1) -> @W2+b2 -> + x
    k_gemm64_wmma<<<gGemm, 128, 0, stream>>>(h, W1, b1, hw, 1);
    k_out_proj  <<<(NN * 4 + T - 1) / T, T, 0, stream>>>(hw, W2, b2, x, out);
}